// Model_76390288327413
// MI455X (gfx1250) — compile-verified
//
#include <hip/hip_runtime.h>

// ---------------------------------------------------------------------------
// Problem constants (from the reference)
// ---------------------------------------------------------------------------
static constexpr int   G_     = 4;
static constexpr int   U_     = 200000;
static constexpr int   I_     = 100000;
static constexpr int   D_     = 64;
static constexpr long long NNZ_ = 1000000;
static constexpr int   LAYERS = 2;
static constexpr float LEAKY  = 0.5f;
static constexpr int   BATCH  = 4096;

typedef __attribute__((ext_vector_type(2))) float v2f;
typedef __attribute__((ext_vector_type(8))) float v8f;

__device__ __forceinline__ float lrelu(float x) { return fmaxf(x, LEAKY * x); }

// ---------------------------------------------------------------------------
// init: cur = emb; gsum += emb   (vectorized float4, n multiple of 4)
// ---------------------------------------------------------------------------
__global__ void init_kernel(const float4* __restrict__ emb,
                            float4* __restrict__ cur,
                            float4* __restrict__ gsum, int n4) {
  int i = blockIdx.x * blockDim.x + threadIdx.x;
  if (i >= n4) return;
  float4 e = emb[i];
  cur[i] = e;
  float4 g = gsum[i];
  g.x += e.x; g.y += e.y; g.z += e.z; g.w += e.w;
  gsum[i] = g;
}

// ---------------------------------------------------------------------------
// propagate: 16 lanes per edge, each lane owns 4 of the 64 columns.
// Gather the source row with global_load_b128, scatter-add into the target
// row with NATIVE f32 atomics (unsafeAtomicAdd -> global_atomic_add_f32;
// a CAS loop here would wreck the dominant kernel). The agg buffers
// (51MB + 26MB) live in the 192MB global L2, so the atomics resolve in L2.
// ---------------------------------------------------------------------------
__global__ __launch_bounds__(256)
void prop_kernel(const float* __restrict__ src_lats,
                 const int* __restrict__ src_idx,
                 const int* __restrict__ tgt_idx,
                 float* __restrict__ agg, int nnz) {
  long long tid = (long long)blockIdx.x * blockDim.x + threadIdx.x;
  int e   = (int)(tid >> 4);   // edge id (16 lanes per edge)
  int sub = (int)(tid & 15);   // lane within edge row
  if (e >= nnz) return;
  if (sub == 0 && e + 4096 < nnz) {
    // stream-prefetch the edge index arrays (global_prefetch_b8 on gfx1250)
    __builtin_prefetch(&src_idx[e + 4096], 0, 0);
    __builtin_prefetch(&tgt_idx[e + 4096], 0, 0);
  }
  int s = src_idx[e];
  int t = tgt_idx[e];
  const float4 v = *(const float4*)(src_lats + (size_t)s * D_ + sub * 4);
  float* dst = agg + (size_t)t * D_ + sub * 4;
  unsafeAtomicAdd(dst + 0, v.x);
  unsafeAtomicAdd(dst + 1, v.y);
  unsafeAtomicAdd(dst + 2, v.z);
  unsafeAtomicAdd(dst + 3, v.w);
}

// ---------------------------------------------------------------------------
// update: cur = act(agg) + cur; gsum += cur   (fused, float4)
// ---------------------------------------------------------------------------
__global__ void update_kernel(float4* __restrict__ cur,
                              const float4* __restrict__ agg,
                              float4* __restrict__ gsum, int n4) {
  int i = blockIdx.x * blockDim.x + threadIdx.x;
  if (i >= n4) return;
  float4 a = agg[i];
  float4 c = cur[i];
  c.x += lrelu(a.x); c.y += lrelu(a.y); c.z += lrelu(a.z); c.w += lrelu(a.w);
  cur[i] = c;
  float4 g = gsum[i];
  g.x += c.x; g.y += c.y; g.z += c.z; g.w += c.w;
  gsum[i] = g;
}

// ---------------------------------------------------------------------------
// Final readout with V_WMMA_F32_16X16X4_F32 (f32 in / f32 acc — exact).
//
//   preds[b] = dot( 0.25*gsum_u[uid[b]] + act(gsum_u[uloc[b]]),
//                   0.25*gsum_i[iid[b]] )          over D=64
//
// One wave handles 16 batch rows. A-frag (16x4 f32, 2 VGPRs): lanes 0-15 hold
// row M=lane, K={0,1}; lanes 16-31 hold row M=lane-16, K={2,3}. B-frag (4x16)
// mirrors this with N in the lane index, so both fragments load the SAME
// (row, k) elements of their respective 16-row panels. Accumulating 16 chunks
// of K=4 covers D=64; the per-row dot products are the diagonal of C.
// C layout: VGPR r, lanes 0-15 -> (M=r, N=lane); lanes 16-31 -> (M=r+8, N=lane-16).
// Diagonal (i,i): i<8 -> lane i, c[i];  i>=8 -> lane i+16 (24..31), c[lane-24].
// ---------------------------------------------------------------------------
__global__ __launch_bounds__(32)
void predict_kernel(const float* __restrict__ gsum_u,
                    const float* __restrict__ gsum_i,
                    const int* __restrict__ uids,
                    const int* __restrict__ iids,
                    const int* __restrict__ ulocs,
                    float* __restrict__ preds) {
  const int lane  = threadIdx.x;        // 0..31, EXEC all-ones (no divergence)
  const int base  = blockIdx.x * 16;
  const int row   = base + (lane & 15);
  const int khalf = lane >> 4;          // 0 or 1

  const float* gu = gsum_u + (size_t)uids[row]  * D_;
  const float* gl = gsum_u + (size_t)ulocs[row] * D_;
  const float* gi = gsum_i + (size_t)iids[row]  * D_;

  v8f c = {};
#pragma unroll
  for (int kc = 0; kc < 16; ++kc) {
    const int k0 = kc * 4 + khalf * 2;
    float2 u0 = *(const float2*)(gu + k0);
    float2 ua = *(const float2*)(gl + k0);
    float2 iv = *(const float2*)(gi + k0);
    v2f a, b;
    a.x = 0.25f * u0.x + lrelu(ua.x);
    a.y = 0.25f * u0.y + lrelu(ua.y);
    b.x = 0.25f * iv.x;
    b.y = 0.25f * iv.y;
    c = __builtin_amdgcn_wmma_f32_16x16x4_f32(
            /*neg_a=*/false, a, /*neg_b=*/false, b,
            /*c_mod=*/(short)0, c, /*reuse_a=*/false, /*reuse_b=*/false);
  }

  float carr[8];
  *(v8f*)carr = c;
  if (lane < 8)        preds[base + lane]        = carr[lane];
  else if (lane >= 24) preds[base + (lane - 16)] = carr[lane - 24];
}

// ---------------------------------------------------------------------------
// Host orchestration
// ---------------------------------------------------------------------------
extern "C" void kernel_launch(void* const* d_in, const int* in_sizes, int n_in,
                              void* d_out, int out_size, void* d_ws, size_t ws_size,
                              hipStream_t stream) {
  (void)in_sizes; (void)n_in; (void)out_size; (void)ws_size;

  const float* ue      = (const float*)d_in[0];  // [G,U,D]
  const float* ie      = (const float*)d_in[1];  // [G,I,D]
  const int*   adj_src = (const int*)d_in[2];    // [G,NNZ] item indices
  const int*   adj_tgt = (const int*)d_in[3];    // [G,NNZ] user indices
  const int*   tp_src  = (const int*)d_in[4];    // [G,NNZ] user indices
  const int*   tp_tgt  = (const int*)d_in[5];    // [G,NNZ] item indices
  const int*   uids    = (const int*)d_in[6];
  const int*   iids    = (const int*)d_in[7];
  const int*   ulocs   = (const int*)d_in[8];
  float*       preds   = (float*)d_out;

  const size_t UD = (size_t)U_ * D_;
  const size_t ID = (size_t)I_ * D_;

  float* ws     = (float*)d_ws;
  float* cur_u  = ws;  ws += UD;
  float* agg_u  = ws;  ws += UD;
  float* gsum_u = ws;  ws += UD;
  float* cur_i  = ws;  ws += ID;
  float* agg_i  = ws;  ws += ID;
  float* gsum_i = ws;  ws += ID;

  hipMemsetAsync(gsum_u, 0, UD * sizeof(float), stream);
  hipMemsetAsync(gsum_i, 0, ID * sizeof(float), stream);

  const int TB = 256;
  const int nU4 = (int)(UD / 4), nI4 = (int)(ID / 4);
  const int gU4 = (nU4 + TB - 1) / TB, gI4 = (nI4 + TB - 1) / TB;
  const long long edge_threads = NNZ_ * 16;   // 16 lanes per edge
  const int gE = (int)((edge_threads + TB - 1) / TB);

  for (int k = 0; k < G_; ++k) {
    init_kernel<<<gU4, TB, 0, stream>>>((const float4*)(ue + (size_t)k * UD),
                                        (float4*)cur_u, (float4*)gsum_u, nU4);
    init_kernel<<<gI4, TB, 0, stream>>>((const float4*)(ie + (size_t)k * ID),
                                        (float4*)cur_i, (float4*)gsum_i, nI4);
    for (int l = 0; l < LAYERS; ++l) {
      hipMemsetAsync(agg_u, 0, UD * sizeof(float), stream);
      hipMemsetAsync(agg_i, 0, ID * sizeof(float), stream);
      // users <- items over subadj
      prop_kernel<<<gE, TB, 0, stream>>>(cur_i, adj_src + (size_t)k * NNZ_,
                                         adj_tgt + (size_t)k * NNZ_, agg_u, (int)NNZ_);
      // items <- users over subtpadj
      prop_kernel<<<gE, TB, 0, stream>>>(cur_u, tp_src + (size_t)k * NNZ_,
                                         tp_tgt + (size_t)k * NNZ_, agg_i, (int)NNZ_);
      update_kernel<<<gU4, TB, 0, stream>>>((float4*)cur_u, (const float4*)agg_u,
                                            (float4*)gsum_u, nU4);
      update_kernel<<<gI4, TB, 0, stream>>>((float4*)cur_i, (const float4*)agg_i,
                                            (float4*)gsum_i, nI4);
    }
  }

  predict_kernel<<<BATCH / 16, 32, 0, stream>>>(gsum_u, gsum_i, uids, iids, ulocs, preds);
}